// HiddenStateMemory_19911468384424
// MI455X (gfx1250) — compile-verified
//
#include <hip/hip_runtime.h>
#include <hip/hip_bf16.h>
#include <math.h>

// Problem constants (from reference setup_inputs)
#define BB      16
#define SS      8192
#define DD      2048
#define STRIDEE 32
#define MM      256     // SS / STRIDEE
#define TOPK    4
#define EPSF    1e-12f

#define KWAVES  8                 // split-K waves per 16-row tile
#define KSEG    (DD / KWAVES)     // 256 K-elements per wave

typedef __attribute__((ext_vector_type(2))) float v2f;
typedef __attribute__((ext_vector_type(8))) float v8f;

// ---------------------------------------------------------------------------
// Kernel 0: per-batch query L2 norm (clamped) -> qn[b]
// ---------------------------------------------------------------------------
__global__ __launch_bounds__(256)
void hsm_qnorm_kernel(const float* __restrict__ q, float* __restrict__ qn) {
    __shared__ float red[256];
    const int b = blockIdx.x, t = threadIdx.x;
    const float* qb = q + (size_t)b * DD;
    float ss = 0.f;
    #pragma unroll
    for (int j = 0; j < DD / 256; ++j) {
        float v = qb[t + j * 256];
        ss = fmaf(v, v, ss);
    }
    red[t] = ss;
    __syncthreads();
    for (int s = 128; s > 0; s >>= 1) {
        if (t < s) red[t] += red[t + s];
        __syncthreads();
    }
    if (t == 0) qn[b] = fmaxf(sqrtf(red[0]), EPSF);
}

// ---------------------------------------------------------------------------
// Kernel 1: cosine scores via V_WMMA_F32_16X16X4_F32 with split-K.
// Block = 256 threads = 8 waves, all working one 16-row tile; wave w owns
// K-range [w*256, w*256+256). Per K-chunk of 4 (ISA 32-bit 16x4/4x16 layout):
//   A (16x4): lane l<16 -> row (base+l) elems {k,k+1}; lane l>=16 -> same
//             row set, elems {k+2,k+3}                       (b64 loads)
//   B (4x16): q chunk broadcast across N (same K-half split) (b64 loads)
// Every column of D accumulates dot(row, q); per-lane sum-of-squares rides
// in VALU (co-executes with the XDL WMMA). Per-wave partials are combined
// across the 8 waves through LDS in a FIXED order (deterministic).
// Grid = 16 batches * 16 tiles = 256 blocks -> 2048 waves in flight.
// ---------------------------------------------------------------------------
__global__ __launch_bounds__(256)
void hsm_scores_wmma_kernel(const float* __restrict__ hs,
                            const unsigned char* __restrict__ mask,  // JAX bool = u8
                            const float* __restrict__ q,
                            const float* __restrict__ qn,
                            float* __restrict__ scores) {
    __shared__ float sdot[KWAVES][16];
    __shared__ float ssumsq[KWAVES][16];

    const int b    = blockIdx.x >> 4;   // 16 tiles per batch
    const int tile = blockIdx.x & 15;   // 16 rows per tile
    const int wave = threadIdx.x >> 5;  // K-segment owner
    const int lane = threadIdx.x & 31;
    const int kh   = lane >> 4;         // K-half: 0 -> {k,k+1}; 1 -> {k+2,k+3}
    const int lrow = lane & 15;
    const int rowbase = tile * 16;
    const int m = rowbase + lrow;       // memory row this lane streams for A
    const int kbase = wave * KSEG;

    const float* arow = hs + ((size_t)b * SS + (size_t)m * STRIDEE) * DD + kbase + 2 * kh;
    const float* qrow = q + (size_t)b * DD + kbase + 2 * kh;

    v8f  c  = {};
    float ss = 0.f;

    #pragma unroll 8
    for (int k = 0; k < KSEG; k += 4) {
        v2f a  = *(const v2f*)(arow + k);   // global_load_b64 (streams 32MB total)
        v2f bq = *(const v2f*)(qrow + k);   // global_load_b64 (L1-hot)
        ss = fmaf(a.x, a.x, ss);
        ss = fmaf(a.y, a.y, ss);
        c = __builtin_amdgcn_wmma_f32_16x16x4_f32(
                /*neg_a=*/false, a, /*neg_b=*/false, bq,
                /*c_mod=*/(short)0, c, /*reuse_a=*/false, /*reuse_b=*/false);
    }

    // Combine K-halves of sum-of-squares within the wave: lanes 0..15 then
    // hold this wave's partial ||row (rowbase+lane)||^2.
    ss += __shfl_xor(ss, 16, 32);

    // Extract D[:,0]: lane 0 holds rows 0..7 in c[0..7]; lane 16 rows 8..15.
    float dot = 0.f;
    #pragma unroll
    for (int r = 0; r < 8; ++r) {
        float lo = __shfl(c[r], 0, 32);    // D[r][0]   -> row rowbase+r
        float hi = __shfl(c[r], 16, 32);   // D[8+r][0] -> row rowbase+8+r
        if (lane == r)     dot = lo;
        if (lane == r + 8) dot = hi;
    }

    if (lane < 16) {
        sdot[wave][lane]   = dot;
        ssumsq[wave][lane] = ss;
    }
    __syncthreads();

    // Fixed-order cross-wave reduction (deterministic), then finalize.
    if (threadIdx.x < 16) {
        const int t = threadIdx.x;
        float dsum = 0.f, nsum = 0.f;
        #pragma unroll
        for (int w = 0; w < KWAVES; ++w) {
            dsum += sdot[w][t];
            nsum += ssumsq[w][t];
        }
        const int mo = rowbase + t;
        float nm = fmaxf(sqrtf(nsum), EPSF);
        float sc = dsum / (nm * qn[b]);
        if (!mask[(size_t)b * SS + (size_t)mo * STRIDEE]) sc = -__builtin_huge_valf();
        scores[b * MM + mo] = sc;
    }
}

// ---------------------------------------------------------------------------
// Kernel 2: per-batch top-4 (argmax x4, tie -> lower index), softmax over the
// 4 scores, gather the raw memory rows (L2-hot), weighted sum -> retrieved;
// emit indices (as float values into the f32 output buffer).
// ---------------------------------------------------------------------------
__global__ __launch_bounds__(256)
void hsm_topk_gather_kernel(const float* __restrict__ hs,
                            const float* __restrict__ scores,
                            float* __restrict__ out_retr,
                            float* __restrict__ out_idx) {
    __shared__ float ls[256];
    __shared__ int   li[256];
    __shared__ float top_s[TOPK];
    __shared__ int   top_i[TOPK];

    const int b = blockIdx.x, t = threadIdx.x;
    float cur = scores[b * MM + t];

    for (int kk = 0; kk < TOPK; ++kk) {
        ls[t] = cur; li[t] = t;
        __syncthreads();
        for (int s = 128; s > 0; s >>= 1) {
            if (t < s) {
                float s2 = ls[t + s]; int i2 = li[t + s];
                if (s2 > ls[t] || (s2 == ls[t] && i2 < li[t])) { ls[t] = s2; li[t] = i2; }
            }
            __syncthreads();
        }
        if (t == 0) { top_s[kk] = ls[0]; top_i[kk] = li[0]; }
        __syncthreads();
        if (t == top_i[kk]) cur = -__builtin_huge_valf();  // remove winner
        __syncthreads();
    }

    // Softmax over the 4 top scores (redundant per thread; registers only).
    float w0, w1, w2, w3;
    {
        float mx = top_s[0];  // top_s sorted descending
        w0 = expf(top_s[0] - mx);
        w1 = expf(top_s[1] - mx);
        w2 = expf(top_s[2] - mx);
        w3 = expf(top_s[3] - mx);
        float inv = 1.0f / (w0 + w1 + w2 + w3);
        w0 *= inv; w1 *= inv; w2 *= inv; w3 *= inv;
    }

    const float* r0 = hs + ((size_t)b * SS + (size_t)top_i[0] * STRIDEE) * DD;
    const float* r1 = hs + ((size_t)b * SS + (size_t)top_i[1] * STRIDEE) * DD;
    const float* r2 = hs + ((size_t)b * SS + (size_t)top_i[2] * STRIDEE) * DD;
    const float* r3 = hs + ((size_t)b * SS + (size_t)top_i[3] * STRIDEE) * DD;

    #pragma unroll
    for (int j = 0; j < DD / 256; ++j) {
        int d = t + j * 256;
        float acc = w0 * r0[d];
        acc = fmaf(w1, r1[d], acc);
        acc = fmaf(w2, r2[d], acc);
        acc = fmaf(w3, r3[d], acc);
        out_retr[(size_t)b * DD + d] = acc;
    }

    if (t < TOPK) out_idx[b * TOPK + t] = (float)top_i[t];
}

// ---------------------------------------------------------------------------
// Launch
//   d_in[0] = hidden_states  f32  (16*8192*2048)
//   d_in[1] = attention_mask bool (16*8192) -> u8
//   d_in[2] = query          f32  (16*2048)
//   d_out   = [retrieved: 16*2048 f32][top_indices: 16*4 as f32]
//   d_ws    = [qn: 16 f32][scores: 16*256 f32]
// ---------------------------------------------------------------------------
extern "C" void kernel_launch(void* const* d_in, const int* in_sizes, int n_in,
                              void* d_out, int out_size, void* d_ws, size_t ws_size,
                              hipStream_t stream) {
    (void)in_sizes; (void)n_in; (void)out_size; (void)ws_size;
    const float*         hs   = (const float*)d_in[0];
    const unsigned char* mask = (const unsigned char*)d_in[1];
    const float*         q    = (const float*)d_in[2];

    float* wsf    = (float*)d_ws;
    float* qn     = wsf;         // 16
    float* scores = wsf + 16;    // 16*256

    float* out_retr = (float*)d_out;
    float* out_idx  = out_retr + (size_t)BB * DD;

    hsm_qnorm_kernel<<<BB, 256, 0, stream>>>(q, qn);
    hsm_scores_wmma_kernel<<<BB * (MM / 16), 256, 0, stream>>>(hs, mask, q, qn, scores);
    hsm_topk_gather_kernel<<<BB, 256, 0, stream>>>(hs, scores, out_retr, out_idx);
}